// ProdAttention_1580547973475
// MI455X (gfx1250) — compile-verified
//
#include <hip/hip_runtime.h>
#include <math.h>

// Problem constants (from reference)
#define B_      32
#define T_      4096
#define D_      1024
#define NCHUNK  8                  // T-chunks per batch -> 32*8 = 256 blocks
#define CHUNK   (T_ / NCHUNK)      // 512 rows per block
#define TILE    16                 // rows per WMMA tile
#define NTILES  (CHUNK / TILE)     // 32 tiles per block
#define NTHREADS 256               // 8 wave32s
#define NWAVES  8
#define DPW     (D_ / NWAVES)      // 128 feature columns per wave (stage 1)
#define K_ITERS (DPW / 4)          // 32 WMMAs (K=4 each) per tile per wave

// LDS tile row stride (floats). 1028 % 64 == 4 makes the stage-1 ds_load_b64
// pattern (lane r reads row r, K-half offset 0/2) hit 64 distinct banks.
#define LDS_STRIDE 1028

typedef __attribute__((ext_vector_type(2))) float v2f;
typedef __attribute__((ext_vector_type(8))) float v8f;
typedef __attribute__((ext_vector_type(4))) int   v4i_t;

// Pointer types matching the async-to-LDS builtin's parameters:
//   (int4 addrspace(1)* gsrc, int4 addrspace(3)* ldst, imm offset, imm cpol)
typedef __attribute__((address_space(1))) v4i_t gv4i_t;
typedef __attribute__((address_space(3))) v4i_t lv4i_t;

// ---- gfx1250 async global->LDS path (guarded; sync fallback keeps compile green)
#if defined(__has_builtin)
#  if __has_builtin(__builtin_amdgcn_global_load_async_to_lds_b128)
#    define HAS_ASYNC_LDS 1
#  endif
#endif
#ifndef HAS_ASYNC_LDS
#  define HAS_ASYNC_LDS 0
#endif

#if defined(__has_builtin) && __has_builtin(__builtin_amdgcn_s_wait_asynccnt)
#  define WAIT_ASYNC(n) __builtin_amdgcn_s_wait_asynccnt(n)
#elif HAS_ASYNC_LDS
#  define WAIT_ASYNC(n) asm volatile("s_wait_asynccnt %0" :: "i"(n) : "memory")
#else
#  define WAIT_ASYNC(n) ((void)0)
#endif

// Stage one 16 x 1024 f32 tile (64 KB) into LDS, 16 B per lane, fully coalesced.
__device__ __forceinline__
void issue_tile_load(const float* __restrict__ gsrc, float* lbuf, int tid)
{
#if HAS_ASYNC_LDS
#pragma unroll
    for (int r = 0; r < TILE; ++r) {
        __builtin_amdgcn_global_load_async_to_lds_b128(
            (gv4i_t*)(gsrc + (size_t)r * D_ + tid * 4),
            (lv4i_t*)(lbuf + r * LDS_STRIDE + tid * 4),
            /*offset=*/0, /*cpol=*/0);
    }
#else
#pragma unroll
    for (int r = 0; r < TILE; ++r) {
        const float4 v = *(const float4*)(gsrc + (size_t)r * D_ + tid * 4);
        *(float4*)(lbuf + r * LDS_STRIDE + tid * 4) = v;
    }
#endif
}

// ---------------------------------------------------------------------------
// Kernel 1: per (batch, chunk) block -- single structural HBM pass over eh.
//   prefetch: async-copy next 16x1024 tile into LDS (double buffered)
//   stage 1 : pax[t] = eh[t,:] . dhx  via V_WMMA_F32_16X16X4_F32 (A from LDS)
//   stage 2 : online softmax + context accumulation (float4 from LDS)
// ---------------------------------------------------------------------------
__global__ __launch_bounds__(NTHREADS)
void attn_partial_kernel(const float* __restrict__ eh,
                         const float* __restrict__ dhx,
                         float* __restrict__ pax_out,   // [B,T] (ax slot of d_out)
                         float* __restrict__ part_c,    // [B,NCHUNK,D]
                         float* __restrict__ part_m,    // [B,NCHUNK]
                         float* __restrict__ part_s)    // [B,NCHUNK]
{
    const int chunk = blockIdx.x;
    const int b     = blockIdx.y;
    const int tid   = threadIdx.x;
    const int lane  = tid & 31;
    const int wv    = tid >> 5;

    __shared__ float ebuf[2][TILE * LDS_STRIDE];  // double-buffered eh tiles (~132 KB)
    __shared__ float red[NWAVES][TILE];           // per-wave partial dot results
    __shared__ float pax_tile[TILE];              // reduced scores for current tile

    // Stage-1 WMMA operand addressing (A: 16x4 f32 layout):
    //   lanes 0-15  hold rows M=0..15 with {K=0,K=1} in VGPR0/1
    //   lanes 16-31 hold rows M=0..15 with {K=2,K=3} in VGPR0/1
    const int dbase = wv * DPW;
    const int koff  = (lane >> 4) << 1;    // 0 or 2
    const int mrow  = lane & 15;

    const size_t eh_b = (size_t)b * T_ * D_;
    const float* dptr = dhx + (size_t)b * D_ + dbase + koff;  // tiny, L0-resident

    // Stage-2 context accumulator: each thread owns 4 consecutive d columns.
    const int dcol = tid * 4;
    float4 c4 = make_float4(0.f, 0.f, 0.f, 0.f);
    float  m_run = -INFINITY;
    float  s_run = 0.f;

    const int row0 = chunk * CHUNK;

    // Preload tile 0.
    issue_tile_load(eh + eh_b + (size_t)row0 * D_, ebuf[0], tid);

    for (int tile = 0; tile < NTILES; ++tile) {
        // Kick off next tile's async copy, then wait for the current tile
        // (async loads complete in order: cnt<=16 -> current tile resident).
        if (tile + 1 < NTILES) {
            issue_tile_load(eh + eh_b + (size_t)(row0 + (tile + 1) * TILE) * D_,
                            ebuf[(tile + 1) & 1], tid);
            WAIT_ASYNC(16);
        } else {
            WAIT_ASYNC(0);
        }
        __syncthreads();
        const float* buf = ebuf[tile & 1];
        const int t0 = row0 + tile * TILE;

        // ---------------- stage 1: WMMA dot products (A from LDS) ----------------
        const float* arow = buf + mrow * LDS_STRIDE + dbase + koff;
        v8f acc = {0.f, 0.f, 0.f, 0.f, 0.f, 0.f, 0.f, 0.f};
#pragma unroll
        for (int i = 0; i < K_ITERS; ++i) {
            v2f a  = *(const v2f*)(arow + 4 * i);          // ds_load_b64, conflict-free
            v2f bv = *(const v2f*)(dptr + 4 * i);          // dhx broadcast over N
            acc = __builtin_amdgcn_wmma_f32_16x16x4_f32(
                    /*neg_a=*/false, a, /*neg_b=*/false, bv,
                    /*c_mod=*/(short)0, acc, /*reuse_a=*/false, /*reuse_b=*/false);
        }
        // C layout: VGPR v -> M=v (lanes 0-15), M=v+8 (lanes 16-31); columns equal.
        if (mrow == 0) {
            const int ro = (lane >> 4) * 8;
#pragma unroll
            for (int v = 0; v < 8; ++v) red[wv][ro + v] = acc[v];
        }
        __syncthreads();

        if (tid < TILE) {
            float p = 0.f;
#pragma unroll
            for (int w = 0; w < NWAVES; ++w) p += red[w][tid];
            pax_tile[tid] = p;
            pax_out[(size_t)b * T_ + t0 + tid] = p;   // raw score, finalized later
        }
        __syncthreads();

        // -------- online softmax update (replicated, identical per thread) ----
        float tmax = pax_tile[0];
#pragma unroll
        for (int t = 1; t < TILE; ++t) tmax = fmaxf(tmax, pax_tile[t]);
        const float new_m = fmaxf(m_run, tmax);
        const float scale = __expf(m_run - new_m);
        c4.x *= scale; c4.y *= scale; c4.z *= scale; c4.w *= scale;
        s_run *= scale;

        // ---------------- stage 2: weighted context accumulation (from LDS) ------
        const float* crow = buf + dcol;
        float s_loc = 0.f;
#pragma unroll
        for (int t = 0; t < TILE; ++t) {
            const float wt = __expf(pax_tile[t] - new_m);
            s_loc += wt;
            const float4 v = *(const float4*)(crow + t * LDS_STRIDE);
            c4.x = fmaf(wt, v.x, c4.x);
            c4.y = fmaf(wt, v.y, c4.y);
            c4.z = fmaf(wt, v.z, c4.z);
            c4.w = fmaf(wt, v.w, c4.w);
        }
        s_run += s_loc;
        m_run  = new_m;
        __syncthreads();   // all reads of buf done before next iter overwrites it
    }

    // ---------------- write per-chunk partials ----------------
    const size_t pco = ((size_t)b * NCHUNK + chunk) * D_ + dcol;
    *(float4*)(part_c + pco) = c4;
    if (tid == 0) {
        part_m[b * NCHUNK + chunk] = m_run;
        part_s[b * NCHUNK + chunk] = s_run;
    }
}

// ---------------------------------------------------------------------------
// Kernel 2: merge chunk partials per batch; finalize sx and ax.
// ---------------------------------------------------------------------------
__global__ __launch_bounds__(NTHREADS)
void attn_combine_kernel(const float* __restrict__ part_c,
                         const float* __restrict__ part_m,
                         const float* __restrict__ part_s,
                         float* __restrict__ sx_out,   // [B,D]
                         float* __restrict__ ax_out)   // [B,T], holds raw pax on entry
{
    const int b   = blockIdx.x;
    const int tid = threadIdx.x;

    float m[NCHUNK], s[NCHUNK], f[NCHUNK];
    float M = -INFINITY;
#pragma unroll
    for (int i = 0; i < NCHUNK; ++i) {
        m[i] = part_m[b * NCHUNK + i];
        s[i] = part_s[b * NCHUNK + i];
        M = fmaxf(M, m[i]);
    }
    float S = 0.f;
#pragma unroll
    for (int i = 0; i < NCHUNK; ++i) {
        f[i] = __expf(m[i] - M);
        S += s[i] * f[i];
    }
    const float invS = 1.0f / S;

    // sx[b, d] = (sum_i c_i[d] * exp(m_i - M)) / S
    const int dcol = tid * 4;
    float4 acc = make_float4(0.f, 0.f, 0.f, 0.f);
#pragma unroll
    for (int i = 0; i < NCHUNK; ++i) {
        const float4 v = *(const float4*)(part_c + ((size_t)b * NCHUNK + i) * D_ + dcol);
        acc.x = fmaf(f[i], v.x, acc.x);
        acc.y = fmaf(f[i], v.y, acc.y);
        acc.z = fmaf(f[i], v.z, acc.z);
        acc.w = fmaf(f[i], v.w, acc.w);
    }
    float4 r;
    r.x = acc.x * invS; r.y = acc.y * invS; r.z = acc.z * invS; r.w = acc.w * invS;
    *(float4*)(sx_out + (size_t)b * D_ + dcol) = r;

    // ax[b, t] = exp(pax - M) / S   (in place over the raw scores)
#pragma unroll
    for (int rr = 0; rr < T_ / NTHREADS; ++rr) {
        const size_t idx = (size_t)b * T_ + (size_t)rr * NTHREADS + tid;
        ax_out[idx] = __expf(ax_out[idx] - M) * invS;
    }
}

// ---------------------------------------------------------------------------
extern "C" void kernel_launch(void* const* d_in, const int* in_sizes, int n_in,
                              void* d_out, int out_size, void* d_ws, size_t ws_size,
                              hipStream_t stream) {
    (void)in_sizes; (void)n_in; (void)out_size; (void)ws_size;

    const float* eh  = (const float*)d_in[0];   // [B,T,D] fp32
    const float* dhx = (const float*)d_in[1];   // [B,1,D] fp32

    float* out    = (float*)d_out;
    float* sx_out = out;                        // [B,D]   = 32768 floats
    float* ax_out = out + (size_t)B_ * D_;      // [B,T]   = 131072 floats

    float* ws     = (float*)d_ws;               // ~1.05 MB used
    float* part_c = ws;                                    // B*NCHUNK*D
    float* part_m = part_c + (size_t)B_ * NCHUNK * D_;     // B*NCHUNK
    float* part_s = part_m + B_ * NCHUNK;                  // B*NCHUNK

    dim3 g1(NCHUNK, B_);
    attn_partial_kernel<<<g1, NTHREADS, 0, stream>>>(eh, dhx, ax_out,
                                                     part_c, part_m, part_s);
    attn_combine_kernel<<<B_, NTHREADS, 0, stream>>>(part_c, part_m, part_s,
                                                     sx_out, ax_out);
}